// CentroidHead_712964571644
// MI455X (gfx1250) — compile-verified
//
#include <hip/hip_runtime.h>
#include <hip/hip_bf16.h>
#include <math.h>

typedef float v2f __attribute__((ext_vector_type(2)));
typedef float v8f __attribute__((ext_vector_type(8)));

#define LATENT    128
#define GDIM      512
#define DDIM      (GDIM + 2)      // 514
#define DD2       (DDIM * DDIM)   // 264196
#define TAU_F     0.1f
#define EMPTY_KEY (-1)

__device__ __forceinline__ unsigned hash_key(int key, unsigned mask) {
  return (((unsigned)key) * 2654435761u) & mask;
}

__device__ __forceinline__ float sigmoid_f(float x) {
  return 1.0f / (1.0f + expf(-x));
}

// ---------------- K1: init hash table keys ----------------
__global__ void init_hash_kernel(int* __restrict__ tkeys, unsigned tsize) {
  unsigned i = blockIdx.x * blockDim.x + threadIdx.x;
  if (i < tsize) tkeys[i] = EMPTY_KEY;
}

// ---------------- K2: logits = feats @ weight + bias via f32 WMMA ----------------
// One wave computes 16 rows. A tile = 16x4 f32 per step (32 steps over K=128).
// A layout (ISA 7.12.2, 32-bit A 16x4): lanes 0-15 hold M=0..15 with K={0,1} in
// the two A VGPRs; lanes 16-31 hold the same M rows with K={2,3}.
// B replicates weight across all 16 columns, so every column of C is the dot
// product; only B's K mapping matters (mirrored from A's layout).
__global__ void gemv_wmma_kernel(const float* __restrict__ feats,
                                 const float* __restrict__ weight,
                                 const float* __restrict__ bias,
                                 float* __restrict__ logits, int n) {
  const int lane    = threadIdx.x & 31;
  const int wave    = threadIdx.x >> 5;
  const int rowBase = (blockIdx.x * (blockDim.x >> 5) + wave) * 16;
  if (rowBase >= n) return;                 // wave-uniform: EXEC stays all-ones

  const int half = lane >> 4;               // 0 => K pair {0,1}; 1 => {2,3}
  const int mrow = lane & 15;               // M index carried by this lane
  const float* arow = feats + (size_t)(rowBase + mrow) * LATENT + half * 2;

  v8f c = {};
  #pragma unroll 8
  for (int k = 0; k < LATENT; k += 4) {
    v2f a = *(const v2f*)(arow + k);        // global_load_b64, 8B aligned
    v2f b;
    b.x = weight[k + half * 2 + 0];
    b.y = weight[k + half * 2 + 1];
    // 8 args: (neg_a, A, neg_b, B, c_mod, C, reuse_a, reuse_b)
    c = __builtin_amdgcn_wmma_f32_16x16x4_f32(false, a, false, b,
                                              (short)0, c, false, false);
  }

  const float bs = bias[0];
  // Column N=0 of C: lane 0 holds M=0..7 in c[0..7], lane 16 holds M=8..15.
  if (mrow == 0) {
    #pragma unroll
    for (int m = 0; m < 8; ++m) {
      int row = rowBase + half * 8 + m;
      logits[row] = c[m] + bs;
    }
  }
}

// ---------------- K3: insert masked points into hash ----------------
__global__ void build_hash_kernel(const float* __restrict__ logits,
                                  const int4* __restrict__ coords,
                                  int* __restrict__ tkeys,
                                  float* __restrict__ theat,
                                  unsigned mask, int n) {
  int i = blockIdx.x * blockDim.x + threadIdx.x;
  if (i >= n) return;
  float heat = sigmoid_f(logits[i]);
  if (!(heat > TAU_F)) return;              // unmasked points contribute -inf: skip
  int4 cdi = coords[i];
  int key = ((cdi.x * DDIM + (cdi.y + 1)) * DDIM + (cdi.z + 1)) * DDIM + (cdi.w + 1);
  unsigned slot = hash_key(key, mask);
  for (;;) {
    int prev = atomicCAS(&tkeys[slot], EMPTY_KEY, key);
    if (prev == EMPTY_KEY || prev == key) { theat[slot] = heat; break; }
    slot = (slot + 1) & mask;
  }
}

// ---------------- K4: 7-point stencil max via hash lookups ----------------
__global__ void peak_kernel(const float* __restrict__ logits,
                            const int4* __restrict__ coords,
                            const int* __restrict__ tkeys,
                            const float* __restrict__ theat,
                            float* __restrict__ out_peak,
                            unsigned mask, int n) {
  int i = blockIdx.x * blockDim.x + threadIdx.x;
  if (i >= n) return;
  float heat = sigmoid_f(logits[i]);
  float result = 0.0f;
  if (heat > TAU_F) {
    int4 cdi = coords[i];
    int key = ((cdi.x * DDIM + (cdi.y + 1)) * DDIM + (cdi.z + 1)) * DDIM + (cdi.w + 1);
    const int deltas[7] = {0, 1, -1, DDIM, -DDIM, DD2, -DD2};
    float maxi = -INFINITY;
    #pragma unroll
    for (int o = 0; o < 7; ++o) {
      int nkey = key + deltas[o];
      unsigned slot = hash_key(nkey, mask);
      for (;;) {
        int k = tkeys[slot];
        if (k == nkey) { maxi = fmaxf(maxi, theat[slot]); break; }
        if (k == EMPTY_KEY) break;
        slot = (slot + 1) & mask;
      }
    }
    result = (heat >= maxi) ? 1.0f : 0.0f;
  }
  out_peak[i] = result;
}

extern "C" void kernel_launch(void* const* d_in, const int* in_sizes, int n_in,
                              void* d_out, int out_size, void* d_ws, size_t ws_size,
                              hipStream_t stream) {
  const float* feats  = (const float*)d_in[0];
  const int4*  coords = (const int4*)d_in[1];
  const float* weight = (const float*)d_in[2];
  const float* bias   = (const float*)d_in[3];
  const int n = in_sizes[0] / LATENT;

  float* logits = (float*)d_out;
  float* peaks  = (float*)d_out + n;

  // Hash table: power-of-two entries, 8 B/entry (key + heat), target ~2N.
  unsigned tsize = 1u << 21;
  while ((size_t)tsize * 8ull > ws_size && tsize > 1024u) tsize >>= 1;
  while ((size_t)tsize * 2ull * 8ull <= ws_size && tsize < (1u << 22)) tsize <<= 1;
  const unsigned mask = tsize - 1u;
  int*   tkeys = (int*)d_ws;
  float* theat = (float*)((char*)d_ws + (size_t)tsize * sizeof(int));

  init_hash_kernel<<<(tsize + 255) / 256, 256, 0, stream>>>(tkeys, tsize);

  const int wavesPerBlock = 8;                 // 256 threads = 8 wave32
  const int rowsPerBlock  = wavesPerBlock * 16;
  gemv_wmma_kernel<<<(n + rowsPerBlock - 1) / rowsPerBlock, wavesPerBlock * 32, 0, stream>>>(
      feats, weight, bias, logits, n);

  build_hash_kernel<<<(n + 255) / 256, 256, 0, stream>>>(logits, coords, tkeys, theat, mask, n);
  peak_kernel<<<(n + 255) / 256, 256, 0, stream>>>(logits, coords, tkeys, theat, peaks, mask, n);
}